// TriPlaneVolume_49151605736023
// MI455X (gfx1250) — compile-verified
//
#include <hip/hip_runtime.h>
#include <stdint.h>

#define RES      512
#define NFEAT    32
#define NPLANES  3
#define TEXELS   (RES * RES)                       // 262144
#define PLANE_ELEMS ((size_t)NFEAT * (size_t)TEXELS) // 8388608 floats per plane

typedef float v4f __attribute__((ext_vector_type(4)));

typedef __attribute__((address_space(1))) int* gptr_i32;
typedef __attribute__((address_space(3))) int* lptr_i32;

#if __has_builtin(__builtin_amdgcn_global_load_async_to_lds_b32)
#define HAVE_ASYNC_LDS 1
#else
#define HAVE_ASYNC_LDS 0
#endif

// ---------------------------------------------------------------------------
// Pass 1: transpose planes [P][F][H][W] -> ws [P][H*W][F] so that the 32
// features of one texel are 128 contiguous bytes (one coalesced segment).
// Tile: 32 features x 64 texels staged in LDS (padded pitch 65 -> no bank
// conflicts on either phase). Staged in via the CDNA5 async-to-LDS path.
// Writes to ws use default temporal hint so the 96 MiB transposed grid
// stays resident in the 192 MB L2 for the sample pass.
// ---------------------------------------------------------------------------
__global__ __launch_bounds__(256) void triplane_transpose(
    const float* __restrict__ planes, float* __restrict__ ws) {
  __shared__ float tile[NFEAT * 65];  // 2080 floats, padded pitch

  const int tid   = threadIdx.x;
  const int pl    = blockIdx.x >> 12;          // 4096 tiles per plane
  const int tbase = (blockIdx.x & 4095) * 64;  // texel base of this tile
  const int tx    = tid & 63;                  // texel within tile
  const int f0    = tid >> 6;                  // 0..3, feature phase

  const float* src = planes + (size_t)pl * PLANE_ELEMS;

#if HAVE_ASYNC_LDS
#pragma unroll
  for (int i = 0; i < 8; ++i) {
    const int f = f0 + i * 4;
    const float* g = src + (size_t)f * TEXELS + (size_t)(tbase + tx);
    __builtin_amdgcn_global_load_async_to_lds_b32(
        (gptr_i32)g,
        (lptr_i32)&tile[f * 65 + tx],
        0, 0);
  }
#if __has_builtin(__builtin_amdgcn_s_wait_asynccnt)
  __builtin_amdgcn_s_wait_asynccnt(0);
#else
  asm volatile("s_wait_asynccnt 0" ::: "memory");
#endif
  __syncthreads();
#else
#pragma unroll
  for (int i = 0; i < 8; ++i) {
    const int f = f0 + i * 4;
    tile[f * 65 + tx] =
        __builtin_nontemporal_load(src + (size_t)f * TEXELS + (size_t)(tbase + tx));
  }
  __syncthreads();
#endif

  // Write out texel-major: ws[pl][(tbase+t)*32 + f], fully coalesced.
  float* dst = ws + (size_t)pl * PLANE_ELEMS + (size_t)tbase * NFEAT;
#pragma unroll
  for (int i = 0; i < 8; ++i) {
    const int linear = tid + i * 256;
    const int f = linear & 31;
    const int t = linear >> 5;
    dst[linear] = tile[f * 65 + t];  // lane l reads word l*65+t -> bank l
  }
}

// ---------------------------------------------------------------------------
// Pass 2 (fast path): sample from the transposed layout.
// Wave32: 4 points per wave, 8 lanes per point, lane sub = feature quad.
// Each corner fetch = global_load_b128, one 128B segment per point (L2 hit).
// Output written with non-temporal b128 stores (384 MB write-once stream;
// must not evict the feature grid from L2).
// ---------------------------------------------------------------------------
__global__ __launch_bounds__(256) void triplane_sample_t(
    const float* __restrict__ pts, const float* __restrict__ ws,
    float* __restrict__ out, int n_pts) {
  const int lane = threadIdx.x & 31;
  const int wave = threadIdx.x >> 5;
  const int grp  = lane >> 3;  // point within wave (0..3)
  const int sub  = lane & 7;   // feature quad (0..7)
  const int p    = (blockIdx.x * 8 + wave) * 4 + grp;
  if (p >= n_pts) return;

  const float px = __builtin_nontemporal_load(pts + 3 * (size_t)p + 0);
  const float py = __builtin_nontemporal_load(pts + 3 * (size_t)p + 1);
  const float pz = __builtin_nontemporal_load(pts + 3 * (size_t)p + 2);

  // grid = [[x,z],[x,y],[y,z]]; (2p-1 -> [-1,1] -> *0.5*(R-1)) == p*(R-1)
  const float S = (float)(RES - 1);
  const float us[3] = {px, px, py};
  const float vs[3] = {pz, py, pz};

#pragma unroll
  for (int pl = 0; pl < NPLANES; ++pl) {
    const float gx = fminf(fmaxf(us[pl] * S, 0.0f), S);
    const float gy = fminf(fmaxf(vs[pl] * S, 0.0f), S);
    const float fx0 = floorf(gx);
    const float fy0 = floorf(gy);
    const float wx = gx - fx0;
    const float wy = gy - fy0;
    const int x0 = (int)fx0;
    const int y0 = (int)fy0;
    const int x1 = min(x0 + 1, RES - 1);
    const int y1 = min(y0 + 1, RES - 1);

    const v4f* base = (const v4f*)(ws + (size_t)pl * PLANE_ELEMS);
    const v4f f00 = base[(size_t)(y0 * RES + x0) * 8 + sub];
    const v4f f10 = base[(size_t)(y0 * RES + x1) * 8 + sub];
    const v4f f01 = base[(size_t)(y1 * RES + x0) * 8 + sub];
    const v4f f11 = base[(size_t)(y1 * RES + x1) * 8 + sub];

    const v4f top = f00 + (f10 - f00) * wx;
    const v4f bot = f01 + (f11 - f01) * wx;
    const v4f r   = top + (bot - top) * wy;

    __builtin_nontemporal_store(r, (v4f*)out + (size_t)p * 24 + pl * 8 + sub);
  }
}

// ---------------------------------------------------------------------------
// Fallback: sample directly from original [P][F][H][W] layout (lane=feature,
// one point per wave). Used only if d_ws is too small for the transpose.
// ---------------------------------------------------------------------------
__global__ __launch_bounds__(256) void triplane_sample_strided(
    const float* __restrict__ pts, const float* __restrict__ planes,
    float* __restrict__ out, int n_pts) {
  const int lane = threadIdx.x & 31;
  const int wave = threadIdx.x >> 5;
  const int p = blockIdx.x * 8 + wave;
  if (p >= n_pts) return;

  const float px = pts[3 * (size_t)p + 0];
  const float py = pts[3 * (size_t)p + 1];
  const float pz = pts[3 * (size_t)p + 2];

  const float S = (float)(RES - 1);
  const float us[3] = {px, px, py};
  const float vs[3] = {pz, py, pz};

#pragma unroll
  for (int pl = 0; pl < NPLANES; ++pl) {
    const float gx = fminf(fmaxf(us[pl] * S, 0.0f), S);
    const float gy = fminf(fmaxf(vs[pl] * S, 0.0f), S);
    const float fx0 = floorf(gx);
    const float fy0 = floorf(gy);
    const float wx = gx - fx0;
    const float wy = gy - fy0;
    const int x0 = (int)fx0;
    const int y0 = (int)fy0;
    const int x1 = min(x0 + 1, RES - 1);
    const int y1 = min(y0 + 1, RES - 1);

    const float* base = planes + (size_t)pl * PLANE_ELEMS + (size_t)lane * TEXELS;
    const float f00 = base[(size_t)(y0 * RES + x0)];
    const float f10 = base[(size_t)(y0 * RES + x1)];
    const float f01 = base[(size_t)(y1 * RES + x0)];
    const float f11 = base[(size_t)(y1 * RES + x1)];

    const float top = fmaf(wx, f10 - f00, f00);
    const float bot = fmaf(wx, f11 - f01, f01);
    const float r   = fmaf(wy, bot - top, top);

    __builtin_nontemporal_store(r, out + (size_t)p * 96 + pl * 32 + lane);
  }
}

extern "C" void kernel_launch(void* const* d_in, const int* in_sizes, int n_in,
                              void* d_out, int out_size, void* d_ws, size_t ws_size,
                              hipStream_t stream) {
  const float* pts    = (const float*)d_in[0];
  const float* planes = (const float*)d_in[1];
  float* out = (float*)d_out;
  const int n_pts = in_sizes[0] / 3;  // 1,048,576

  const size_t need = (size_t)NPLANES * PLANE_ELEMS * sizeof(float);  // ~96 MiB

  if (ws_size >= need) {
    float* ws = (float*)d_ws;
    triplane_transpose<<<dim3(NPLANES * 4096), dim3(256), 0, stream>>>(planes, ws);
    const int grid = (n_pts + 31) / 32;  // 32 points per block (8 waves x 4)
    triplane_sample_t<<<dim3(grid), dim3(256), 0, stream>>>(pts, ws, out, n_pts);
  } else {
    const int grid = (n_pts + 7) / 8;  // 8 points per block (1 per wave)
    triplane_sample_strided<<<dim3(grid), dim3(256), 0, stream>>>(pts, planes, out, n_pts);
  }
}